// CommunicationGNN_71683004170821
// MI455X (gfx1250) — compile-verified
//
#include <hip/hip_runtime.h>
#include <hip/hip_bf16.h>

typedef float v2f __attribute__((ext_vector_type(2)));
typedef float v8f __attribute__((ext_vector_type(8)));

#define DD 128
#define HH 4
#define CC 32

// ---------------------------------------------------------------------------
// generic fill (capture-safe replacement for hipMemset)
// ---------------------------------------------------------------------------
__global__ void gnn_fill_u32(unsigned int* __restrict__ p, unsigned int v, long long n) {
    long long i = (long long)blockIdx.x * blockDim.x + threadIdx.x;
    if (i < n) p[i] = v;
}

// ---------------------------------------------------------------------------
// edge MLP + scatter-sum into h_sum, count per dst node
// one block (128 threads) per edge; thread d owns output channel d
// ---------------------------------------------------------------------------
__global__ void gnn_edge_init(const float* __restrict__ x, const int* __restrict__ ei,
                              const float* __restrict__ W_edge, const float* __restrict__ b_edge,
                              float* __restrict__ h_sum, float* __restrict__ cnt, int E_) {
    int e = blockIdx.x;
    int d = threadIdx.x;
    int dstn = ei[E_ + e];                       // col = edge_index[1]
    float x0 = x[e * 3 + 0], x1 = x[e * 3 + 1], x2 = x[e * 3 + 2];
    float v = x0 * W_edge[d] + x1 * W_edge[DD + d] + x2 * W_edge[2 * DD + d] + b_edge[d];
    atomicAdd(&h_sum[(long long)dstn * DD + d], v);
    if (d == 0) atomicAdd(&cnt[dstn], 1.0f);
}

// ---------------------------------------------------------------------------
// fp32 WMMA GEMM: C[M,128] = A[M,128] @ B[128,128] (+bias) (/clip(cnt,1) per row)
// One wave -> one 16x16 tile, K stepped by 4 via V_WMMA_F32_16X16X4_F32.
// A frag (16x4):  lane<16 holds K0=0,1 ; lane>=16 holds K0=2,3 ; M = lane&15
// B frag (4x16):  lane<16 holds K0=0,1 ; lane>=16 holds K0=2,3 ; N = lane&15
// C/D (16x16):    VGPR r: row = r + 8*(lane>=16), col = lane&15
// ---------------------------------------------------------------------------
__global__ void gnn_wmma_gemm128(const float* __restrict__ A, const float* __restrict__ B,
                                 float* __restrict__ C, const float* __restrict__ bias,
                                 const float* __restrict__ cnt, int M) {
    int wave = threadIdx.x >> 5;                 // 0..3 -> n-subtile
    int lane = threadIdx.x & 31;
    int m0 = blockIdx.x * 16;
    int n0 = (blockIdx.y * 4 + wave) * 16;
    int half = lane >> 4;                        // 0|1
    int l    = lane & 15;
    int K0   = half * 2;
    int Mrow = m0 + l;
    if (Mrow >= M) Mrow = M - 1;                 // uniform-safe clamp (EXEC stays full)
    int Ncol = n0 + l;

    const float* Arow = A + (long long)Mrow * DD;
    v8f acc = {};
#pragma unroll
    for (int k = 0; k < DD; k += 4) {
        v2f a = *(const v2f*)(Arow + k + K0);    // 8B-aligned
        v2f b;
        b.x = B[(long long)(k + K0) * DD + Ncol];
        b.y = B[(long long)(k + K0 + 1) * DD + Ncol];
        acc = __builtin_amdgcn_wmma_f32_16x16x4_f32(false, a, false, b,
                                                    (short)0, acc, false, false);
    }
#pragma unroll
    for (int r = 0; r < 8; ++r) {
        int row = m0 + r + half * 8;
        if (row < M) {
            float v = acc[r];
            if (cnt)  v *= 1.0f / fmaxf(cnt[row], 1.0f);   // fused scatter-mean divide
            if (bias) v += bias[n0 + l];
            C[(long long)row * DD + n0 + l] = v;
        }
    }
}

// ---------------------------------------------------------------------------
// per-node attention logits: a_s[n,h] = sum_c xh[n,h,c]*att_src[h,c]  (same a_d)
// wave32: each head's 32 channels are exactly one wave -> shuffle reduce
// ---------------------------------------------------------------------------
__global__ void gnn_att_scores(const float* __restrict__ xh,
                               const float* __restrict__ att_src,
                               const float* __restrict__ att_dst,
                               float* __restrict__ a_s, float* __restrict__ a_d) {
    int n = blockIdx.x;
    int d = threadIdx.x;
    int head = d >> 5;
    float v  = xh[(long long)n * DD + d];
    float s  = v * att_src[d];
    float t  = v * att_dst[d];
#pragma unroll
    for (int m = 16; m; m >>= 1) { s += __shfl_xor(s, m); t += __shfl_xor(t, m); }
    if ((d & 31) == 0) {
        a_s[n * HH + head] = s;
        a_d[n * HH + head] = t;
    }
}

__device__ __forceinline__ void edge_st(int t, int E_, const int* ei, int& s, int& dd) {
    if (t < E_) { s = ei[t]; dd = ei[E_ + t]; }
    else        { s = t - E_; dd = t - E_; }      // self loops appended
}

// pass 1: segment max of leaky-relu'd logits per (dst, head) — int-trick float atomic max
__global__ void gnn_edge_max(const int* __restrict__ ei, const float* __restrict__ a_s,
                             const float* __restrict__ a_d, float* __restrict__ mval,
                             int E_, int Nn) {
    int idx = blockIdx.x * blockDim.x + threadIdx.x;
    int tot = E_ + Nn;
    int t = idx >> 2, hd = idx & 3;
    if (t >= tot) return;
    int s, dd; edge_st(t, E_, ei, s, dd);
    float e = a_s[s * HH + hd] + a_d[dd * HH + hd];
    e = (e > 0.0f) ? e : 0.2f * e;
    int* p = (int*)&mval[dd * HH + hd];
    if (e >= 0.0f) atomicMax(p, __float_as_int(e));
    else           atomicMin((unsigned int*)p, __float_as_uint(e));
}

// pass 2: denom = segment_sum(exp(e - m[dst]))
__global__ void gnn_edge_denom(const int* __restrict__ ei, const float* __restrict__ a_s,
                               const float* __restrict__ a_d, const float* __restrict__ mval,
                               float* __restrict__ denom, int E_, int Nn) {
    int idx = blockIdx.x * blockDim.x + threadIdx.x;
    int tot = E_ + Nn;
    int t = idx >> 2, hd = idx & 3;
    if (t >= tot) return;
    int s, dd; edge_st(t, E_, ei, s, dd);
    float e = a_s[s * HH + hd] + a_d[dd * HH + hd];
    e = (e > 0.0f) ? e : 0.2f * e;
    atomicAdd(&denom[dd * HH + hd], __expf(e - mval[dd * HH + hd]));
}

// pass 3: out[dst] += alpha * xh[src]  — one block (128 threads) per edge
__global__ void gnn_edge_accum(const int* __restrict__ ei, const float* __restrict__ a_s,
                               const float* __restrict__ a_d, const float* __restrict__ mval,
                               const float* __restrict__ denom, const float* __restrict__ xh,
                               float* __restrict__ out, int E_, int Nn) {
    int t = blockIdx.x;
    int d = threadIdx.x;
    int hd = d >> 5;
    int s, dd; edge_st(t, E_, ei, s, dd);
    float e = a_s[s * HH + hd] + a_d[dd * HH + hd];
    e = (e > 0.0f) ? e : 0.2f * e;
    float alpha = __expf(e - mval[dd * HH + hd]) / denom[dd * HH + hd];
    atomicAdd(&out[(long long)dd * DD + d], alpha * xh[(long long)s * DD + d]);
}

// ---------------------------------------------------------------------------
// bias + LayerNorm + (relu) + residual.  D=128 = 4 waves; shuffle + tiny LDS combine.
// ---------------------------------------------------------------------------
__global__ void gnn_finalize(const float* __restrict__ out, const float* __restrict__ gbias,
                             const float* __restrict__ ln_g, const float* __restrict__ ln_b,
                             float* __restrict__ h, int do_relu) {
    __shared__ float red[4];
    int n = blockIdx.x;
    int d = threadIdx.x;
    float v = out[(long long)n * DD + d] + gbias[d];
    float s = v;
#pragma unroll
    for (int m = 16; m; m >>= 1) s += __shfl_xor(s, m);
    if ((d & 31) == 0) red[d >> 5] = s;
    __syncthreads();
    float mean = (red[0] + red[1] + red[2] + red[3]) * (1.0f / DD);
    float c = v - mean;
    float q = c * c;
#pragma unroll
    for (int m = 16; m; m >>= 1) q += __shfl_xor(q, m);
    __syncthreads();
    if ((d & 31) == 0) red[d >> 5] = q;
    __syncthreads();
    float var = (red[0] + red[1] + red[2] + red[3]) * (1.0f / DD);
    float y = c * rsqrtf(var + 1e-5f) * ln_g[d] + ln_b[d];
    if (do_relu) y = fmaxf(y, 0.0f);
    h[(long long)n * DD + d] += y;
}

// ---------------------------------------------------------------------------
extern "C" void kernel_launch(void* const* d_in, const int* in_sizes, int n_in,
                              void* d_out, int out_size, void* d_ws, size_t ws_size,
                              hipStream_t stream) {
    const float* x      = (const float*)d_in[0];
    const int*   ei     = (const int*)d_in[1];
    const float* W_edge = (const float*)d_in[2];
    const float* b_edge = (const float*)d_in[3];
    const float* W_node = (const float*)d_in[4];
    const float* b_node = (const float*)d_in[5];
    const float* lin_w  = (const float*)d_in[6];
    const float* att_sp = (const float*)d_in[7];
    const float* att_dp = (const float*)d_in[8];
    const float* g_bias = (const float*)d_in[9];
    const float* ln_g   = (const float*)d_in[10];
    const float* ln_b   = (const float*)d_in[11];

    const int E_ = in_sizes[0] / 3;            // x is [E,3]
    const int Nn = out_size / DD;              // output is [N,128]
    const int tot = E_ + Nn;

    // workspace layout (h lives in d_out as the residual accumulator)
    float* bufA  = (float*)d_ws;               // h_sum, then reused as GAT out-accum
    float* xh    = bufA + (long long)Nn * DD;
    float* cnt   = xh   + (long long)Nn * DD;
    float* a_s   = cnt  + Nn;
    float* a_d   = a_s  + (long long)Nn * HH;
    float* mval  = a_d  + (long long)Nn * HH;
    float* denom = mval + (long long)Nn * HH;
    float* h     = (float*)d_out;

    long long nh = (long long)Nn * DD;
    int gh  = (int)((nh + 255) / 256);
    int g4  = (Nn * HH + 255) / 256;
    int g1  = (Nn + 255) / 256;
    int geh = (tot * HH + 255) / 256;
    dim3 gemm_grid((Nn + 15) / 16, 2);         // 4 waves/block cover 64 cols

    // ---- scatter-mean init: h = (scatter_sum(x@W_edge)/cnt) @ W_node + b ----
    gnn_fill_u32<<<gh, 256, 0, stream>>>((unsigned int*)bufA, 0u, nh);
    gnn_fill_u32<<<g1, 256, 0, stream>>>((unsigned int*)cnt, 0u, Nn);
    gnn_edge_init<<<E_, DD, 0, stream>>>(x, ei, W_edge, b_edge, bufA, cnt, E_);
    gnn_wmma_gemm128<<<gemm_grid, 128, 0, stream>>>(bufA, W_node, h, b_node, cnt, Nn);

    // ---- 3 GAT layers ----
    for (int i = 0; i < 3; ++i) {
        gnn_wmma_gemm128<<<gemm_grid, 128, 0, stream>>>(
            h, lin_w + (long long)i * DD * DD, xh, nullptr, nullptr, Nn);
        gnn_att_scores<<<Nn, DD, 0, stream>>>(xh, att_sp + i * DD, att_dp + i * DD, a_s, a_d);

        gnn_fill_u32<<<gh, 256, 0, stream>>>((unsigned int*)bufA, 0u, nh);
        gnn_fill_u32<<<g4, 256, 0, stream>>>((unsigned int*)denom, 0u, Nn * HH);
        gnn_fill_u32<<<g4, 256, 0, stream>>>((unsigned int*)mval, 0xFF800000u, Nn * HH); // -inf

        gnn_edge_max  <<<geh, 256, 0, stream>>>(ei, a_s, a_d, mval, E_, Nn);
        gnn_edge_denom<<<geh, 256, 0, stream>>>(ei, a_s, a_d, mval, denom, E_, Nn);
        gnn_edge_accum<<<tot, DD, 0, stream>>>(ei, a_s, a_d, mval, denom, xh, bufA, E_, Nn);

        gnn_finalize<<<Nn, DD, 0, stream>>>(bufA, g_bias + i * DD, ln_g + i * DD,
                                            ln_b + i * DD, h, (i < 2) ? 1 : 0);
    }
    (void)n_in; (void)ws_size; (void)W_edge;
}